// CRF_16836271800945
// MI455X (gfx1250) — compile-verified
//
#include <hip/hip_runtime.h>
#include <stdint.h>

#define B_ 64
#define T_ 1024
#define K_ 256
#define NEG_INF (-3.402823466e38f)

typedef float v2f __attribute__((ext_vector_type(2)));
typedef unsigned int v4u __attribute__((ext_vector_type(4)));
typedef int v4i __attribute__((ext_vector_type(4)));
typedef int v8i __attribute__((ext_vector_type(8)));

// ---------------------------------------------------------------------------
// CDNA5 async-to-LDS helpers (device / gfx1250 only)
// ---------------------------------------------------------------------------
#if defined(__gfx1250__)
#define GFX1250_ASYNC 1
#if __has_builtin(__builtin_amdgcn_tensor_load_to_lds)
#define GFX1250_TDM 1
#endif

__device__ __forceinline__ void async_ld_b32(const float* g, unsigned lds_off) {
  // GLOBAL_LOAD_ASYNC_TO_LDS_B32, GV mode (saddr = off). VDST = per-lane LDS
  // byte offset, VADDR = 64-bit global address. Tracked by ASYNCcnt.
  asm volatile("global_load_async_to_lds_b32 %0, %1, off"
               :: "v"(lds_off), "v"(g)
               : "memory");
}

__device__ __forceinline__ void wait_async_le1() {
#if __has_builtin(__builtin_amdgcn_s_wait_asynccnt)
  __builtin_amdgcn_s_wait_asynccnt(1);
#else
  asm volatile("s_wait_asynccnt 0x1" ::: "memory");
#endif
}
#endif

// ---------------------------------------------------------------------------
// Forward kernel: one block per batch element, 512 threads (16 wave32).
// Thread (h, j): h = tid>>8 in {0,1}, j = tid & 255 = target label k_next.
// Each thread keeps trans[h*128 .. h*128+127][j] as 64 float2 pairs in VGPRs;
// the running state row is broadcast from an LDS double buffer (ds_load_b128).
// Inner op is max-plus restructured for v_pk_add_f32 + v_max3_num_f32:
//   ~1 VALU instruction per lattice point (bit-exact vs scalar add/max).
// Full state rows are checkpointed to d_ws for exact backpointer recompute.
// ---------------------------------------------------------------------------
__global__ __launch_bounds__(512, 1)
void crf_forward(const float* __restrict__ em,
                 const float* __restrict__ trans,
                 float* __restrict__ st) {
  __shared__ __align__(16) float stateLds[2][K_];
  __shared__ float part[512];
#ifdef GFX1250_ASYNC
  __shared__ __align__(16) float emStage[2][K_];
#endif

  const int tid = threadIdx.x;
  const int j   = tid & (K_ - 1);
  const int h   = tid >> 8;          // 0 or 1: which half of k_prev
  const int b   = blockIdx.x;

  // Preload this thread's transition column slice into register pairs.
  v2f w2[64];
#pragma unroll
  for (int i = 0; i < 64; ++i) {
    w2[i].x = trans[(h * 128 + 2 * i + 0) * K_ + j];
    w2[i].y = trans[(h * 128 + 2 * i + 1) * K_ + j];
  }

  // state_0 = emissions[b, 0, :]
  if (h == 0) {
    const float e0 = em[(b * T_) * K_ + j];
    stateLds[0][j] = e0;
    st[b * K_ + j] = e0;
  }
#ifdef GFX1250_ASYNC
  if (h == 0)  // stage emissions row t=1 (async, ASYNCcnt -> 1)
    async_ld_b32(&em[(b * T_ + 1) * K_ + j],
                 (unsigned)(uintptr_t)&emStage[1][j]);
#endif
  __syncthreads();

  for (int t = 1; t < T_; ++t) {
    const int cur = (t - 1) & 1, nxt = t & 1;

#ifdef GFX1250_ASYNC
    // Stage next emissions row while we compute (duplicate last row at the
    // end so exactly 2 async loads are always in flight -> wait<=1 is exact).
    {
      const int tn = (t + 1 < T_) ? (t + 1) : t;
      if (h == 0)
        async_ld_b32(&em[(b * T_ + tn) * K_ + j],
                     (unsigned)(uintptr_t)&emStage[(t + 1) & 1][j]);
    }
    float e_cur = 0.0f;
#else
    float e_cur = (h == 0) ? em[(b * T_ + t) * K_ + j] : 0.0f;
#endif

    // Max-plus over this thread's 128 k_prev values.
    // Per 4 points: 2x v_pk_add_f32 + 2x v_max3_num_f32, 4 independent chains.
    float m0 = NEG_INF, m1 = NEG_INF, m2 = NEG_INF, m3 = NEG_INF;
    const float4* sp = reinterpret_cast<const float4*>(&stateLds[cur][h * 128]);
#pragma unroll
    for (int c = 0; c < 32; ++c) {
      const float4 s = sp[c];                       // ds_load_b128 broadcast
      v2f slo; slo.x = s.x; slo.y = s.y;
      v2f shi; shi.x = s.z; shi.y = s.w;
      const v2f a = slo + w2[2 * c + 0];            // v_pk_add_f32
      const v2f d = shi + w2[2 * c + 1];            // v_pk_add_f32
      if (c & 1) {
        m2 = fmaxf(m2, fmaxf(a.x, a.y));            // v_max3_num_f32
        m3 = fmaxf(m3, fmaxf(d.x, d.y));            // v_max3_num_f32
      } else {
        m0 = fmaxf(m0, fmaxf(a.x, a.y));
        m1 = fmaxf(m1, fmaxf(d.x, d.y));
      }
    }
    part[tid] = fmaxf(fmaxf(m0, m1), fmaxf(m2, m3));
    __syncthreads();

    if (h == 0) {
#ifdef GFX1250_ASYNC
      wait_async_le1();                 // drain row t (row t+1 stays in flight)
      e_cur = emStage[t & 1][j];
#endif
      const float v = fmaxf(part[j], part[j + K_]) + e_cur;
      stateLds[nxt][j] = v;
      st[(t * B_ + b) * K_ + j] = v;    // checkpoint state row (L2-resident)
    }
    __syncthreads();
  }
}

// ---------------------------------------------------------------------------
// Backtrace: one wave per batch element. Recomputes each needed backpointer
// argmax_k(state[t-1][k] + trans[k][tag]) from the checkpointed states.
// The padded LDS transition image (row stride 257 words -> bank = (k+tag)%64,
// conflict-free column gathers) is produced by a single TDM
// tensor_load_to_lds with pad_enable (TENSORcnt), with a plain-copy fallback.
// Butterfly shfl argmax with first-index tie-break matches jnp.argmax.
// ---------------------------------------------------------------------------
__global__ __launch_bounds__(256, 1)
void crf_backtrace(const float* __restrict__ trans,
                   const float* __restrict__ st,
                   float* __restrict__ out) {
  extern __shared__ float transP[];    // [K_][K_+1]
  const int b = blockIdx.x;

#ifdef GFX1250_TDM
  if (threadIdx.x < 32) {              // one wave issues the TDM copy
    const unsigned lds_base = (unsigned)(uintptr_t)&transP[0];
    const unsigned long long ga = (unsigned long long)(uintptr_t)trans;
    // D# group 0: count=1, lds_addr, 57-bit global_addr, type=2 ("image")
    v4u g0;
    g0.x = 1u;
    g0.y = lds_base;
    g0.z = (unsigned)(ga & 0xFFFFFFFFu);
    g0.w = (unsigned)((ga >> 32) & 0x01FFFFFFu) | 0x80000000u;
    // D# group 1: data_size=4B (2), pad_enable=1, pad_interval=7 (256 DW),
    // pad_amount=0 (+1 DW) -> LDS row stride 257 words.
    v8i g1;
    g1[0] = (int)((2u << 16) | (1u << 20) | (7u << 22));
    g1[1] = (int)(256u << 16);   // tensor_dim0 = 256 (lo16)
    g1[2] = (int)(256u << 16);   // tensor_dim0 hi = 0 | tensor_dim1 lo16 = 256
    g1[3] = (int)(256u << 16);   // tensor_dim1 hi = 0 | tile_dim0 = 256
    g1[4] = 256;                 // tile_dim1 = 256, tile_dim2 = 0
    g1[5] = 256;                 // tensor_dim0_stride = 256
    g1[6] = 0;                   // stride0 hi | tensor_dim1_stride lo16
    g1[7] = 1;                   // tensor_dim1_stride = 65536
    v4i z4 = {0, 0, 0, 0};
#if __clang_major__ >= 23
    v8i z8 = {0, 0, 0, 0, 0, 0, 0, 0};
    __builtin_amdgcn_tensor_load_to_lds(g0, g1, z4, z4, z8, 0);
#else
    __builtin_amdgcn_tensor_load_to_lds(g0, g1, z4, z4, 0);
#endif
#if __has_builtin(__builtin_amdgcn_s_wait_tensorcnt)
    __builtin_amdgcn_s_wait_tensorcnt(0);
#else
    asm volatile("s_wait_tensorcnt 0x0" ::: "memory");
#endif
  }
#else
  // Fallback: cooperative padded copy of transitions into LDS.
  for (int idx = threadIdx.x; idx < K_ * K_; idx += 256) {
    const int k = idx >> 8, jj = idx & (K_ - 1);
    transP[k * (K_ + 1) + jj] = trans[idx];
  }
#endif
  __syncthreads();
  if (threadIdx.x >= 32) return;       // single wave continues
  const int lane = threadIdx.x;

  // last_tag = argmax_k state[T-1][b][k] (first max wins ties)
  float bv = NEG_INF; int bk = 0;
  {
    const float* sr = &st[((T_ - 1) * B_ + b) * K_];
#pragma unroll
    for (int i = 0; i < 8; ++i) {
      const int k = lane + 32 * i;     // ascending per lane -> '>' keeps first
      const float v = sr[k];
      if (v > bv) { bv = v; bk = k; }
    }
#pragma unroll
    for (int m = 16; m >= 1; m >>= 1) {
      const float ov = __shfl_xor(bv, m, 32);
      const int   ok = __shfl_xor(bk, m, 32);
      if (ov > bv || (ov == bv && ok < bk)) { bv = ov; bk = ok; }
    }
  }
  int tag = bk;
  out[b * T_ + (T_ - 1)] = (float)tag;

  for (int t = T_ - 1; t >= 1; --t) {
    const float* sr = &st[((t - 1) * B_ + b) * K_];
    if (t >= 2)                        // prefetch next state row (1KB / wave)
      __builtin_prefetch((const void*)((const char*)(sr - B_ * K_) + lane * 32), 0, 0);

    bv = NEG_INF; bk = 0;
#pragma unroll
    for (int i = 0; i < 8; ++i) {
      const int k = lane + 32 * i;
      const float v = sr[k] + transP[k * (K_ + 1) + tag];
      if (v > bv) { bv = v; bk = k; }
    }
#pragma unroll
    for (int m = 16; m >= 1; m >>= 1) {
      const float ov = __shfl_xor(bv, m, 32);
      const int   ok = __shfl_xor(bk, m, 32);
      if (ov > bv || (ov == bv && ok < bk)) { bv = ov; bk = ok; }
    }
    tag = bk;
    out[b * T_ + (t - 1)] = (float)tag;
  }
}

// ---------------------------------------------------------------------------
extern "C" void kernel_launch(void* const* d_in, const int* in_sizes, int n_in,
                              void* d_out, int out_size, void* d_ws, size_t ws_size,
                              hipStream_t stream) {
  (void)in_sizes; (void)n_in; (void)out_size; (void)ws_size;
  const float* em = (const float*)d_in[0];   // [B, T, K]
  const float* tr = (const float*)d_in[1];   // [K, K]
  float* st  = (float*)d_ws;                 // [T, B, K] state checkpoints (64 MiB)
  float* out = (float*)d_out;                // [B, T] tags as float

  crf_forward<<<dim3(B_), dim3(512), 0, stream>>>(em, tr, st);

  const size_t shmem = (size_t)K_ * (K_ + 1) * sizeof(float);  // 263,168 B
  (void)hipFuncSetAttribute(reinterpret_cast<const void*>(crf_backtrace),
                            hipFuncAttributeMaxDynamicSharedMemorySize,
                            (int)shmem);
  crf_backtrace<<<dim3(B_), dim3(256), shmem, stream>>>(tr, st, out);
}